// AttentionLayer_32865089749681
// MI455X (gfx1250) — compile-verified
//
#include <hip/hip_runtime.h>
#include <math.h>

// Problem sizes (fixed by the reference)
#define B_    32
#define S_    4096
#define DIN_  1024
#define DOUT_ 1024

typedef __attribute__((ext_vector_type(2))) float v2f;
typedef __attribute__((ext_vector_type(8))) float v8f;

// ---------------------------------------------------------------------------
// Kernel 1: x[b,o] = sum_i input[b,i] * W[o,i]   via V_WMMA_F32_16X16X4_F32
// One wave (32 threads) computes one 16x16 tile of x. grid = (B/16, DOUT/16).
// A fragment (16x4 f32): lanes 0-15 hold M=lane, K = k+0/k+1 in vgpr0/1;
//                        lanes 16-31 hold M=lane-16, K = k+2/k+3.
// B fragment (4x16 f32): symmetric (N = lane&15, K split by lane>>4).
// C/D (16x16 f32, 8 VGPRs): vgpr r -> M = r + 8*(lane>>4), N = lane&15.
// ---------------------------------------------------------------------------
__global__ void __launch_bounds__(32)
xw_wmma_kernel(const float* __restrict__ input,   // [B, DIN]
               const float* __restrict__ W,       // [DOUT, DIN]
               float* __restrict__ x_out)         // [B, DOUT]
{
    const int lane  = threadIdx.x;            // 0..31, wave32
    const int mbase = blockIdx.x * 16;        // row tile in B
    const int nbase = blockIdx.y * 16;        // col tile in DOUT
    const int half  = lane >> 4;              // 0: K=0,1  1: K=2,3
    const int l15   = lane & 15;

    const float* arow = input + (size_t)(mbase + l15) * DIN_;  // A: M = l15
    const float* brow = W     + (size_t)(nbase + l15) * DIN_;  // B: N = l15 (W[o,i])

    v8f c = {};  // accumulator starts at zero
    #pragma unroll 4
    for (int k = 0; k < DIN_; k += 4) {
        const int kk = k + 2 * half;
        v2f a, b;
        a.x = arow[kk];     a.y = arow[kk + 1];
        b.x = brow[kk];     b.y = brow[kk + 1];
        // 8 args: (neg_a, A, neg_b, B, c_mod, C, reuse_a, reuse_b)
        c = __builtin_amdgcn_wmma_f32_16x16x4_f32(false, a, false, b,
                                                  (short)0, c, false, false);
    }

    float* orow = x_out + (size_t)(mbase + 8 * half) * DOUT_ + (nbase + l15);
    #pragma unroll
    for (int r = 0; r < 8; ++r)
        orow[(size_t)r * DOUT_] = c[r];
}

// ---------------------------------------------------------------------------
// Kernel 2: fused score + online softmax + partial context, one pass over
// source_hids.  grid = (B, S/256); block = 256 threads (8 waves).
// Each wave processes 32 consecutive source rows; each lane owns feature
// indices { j*128 + lane*4 .. +3 : j=0..7 } (coalesced float4 lanes).
// ---------------------------------------------------------------------------
#define CHUNK 256  // S rows per block

__global__ void __launch_bounds__(256)
attn_stream_kernel(const float* __restrict__ source_hids, // [B,S,DOUT]
                   const float* __restrict__ seg_scores,  // [B,S]
                   const float* __restrict__ xw,          // [B,DOUT]
                   float* __restrict__ raw_scores,        // ws: [B,S]
                   float* __restrict__ partM,             // ws: [B*16]
                   float* __restrict__ partL,             // ws: [B*16]
                   float* __restrict__ partCtx)           // ws: [B*16,DOUT]
{
    const int b     = blockIdx.x;
    const int chunk = blockIdx.y;
    const int lane  = threadIdx.x & 31;
    const int wave  = threadIdx.x >> 5;

    __shared__ float sacc[DOUT_];
    __shared__ float wm[8], wl[8];
    for (int i = threadIdx.x; i < DOUT_; i += 256) sacc[i] = 0.0f;

    // Preload this lane's fragment of x[b,:] into registers (constant per row)
    const float4* xv = (const float4*)(xw + (size_t)b * DOUT_);
    float4 xr[8];
    #pragma unroll
    for (int j = 0; j < 8; ++j) xr[j] = xv[j * 32 + lane];

    const float* srcB = source_hids + (size_t)b * S_ * DOUT_;
    const int s0 = chunk * CHUNK + wave * 32;

    float m = -3.0e38f, l = 0.0f;
    float4 acc[8];
    #pragma unroll
    for (int j = 0; j < 8; ++j) acc[j] = make_float4(0.f, 0.f, 0.f, 0.f);

    for (int r = 0; r < 32; ++r) {
        const int s = s0 + r;
        const float4* row = (const float4*)(srcB + (size_t)s * DOUT_);
        if (r + 1 < 32)  // hint next row toward the caches
            __builtin_prefetch(srcB + (size_t)(s + 1) * DOUT_ + lane * 4, 0, 3);

        float4 v[8];
        #pragma unroll
        for (int j = 0; j < 8; ++j) v[j] = row[j * 32 + lane];

        float p = 0.0f;
        #pragma unroll
        for (int j = 0; j < 8; ++j)
            p += v[j].x * xr[j].x + v[j].y * xr[j].y +
                 v[j].z * xr[j].z + v[j].w * xr[j].w;
        #pragma unroll
        for (int off = 16; off >= 1; off >>= 1)
            p += __shfl_xor(p, off, 32);           // all lanes get the sum

        const float seg = seg_scores[(size_t)b * S_ + s];
        const float t   = p - 100.0f * (1.0f - seg);
        if (lane == 0) raw_scores[(size_t)b * S_ + s] = t;

        // Online softmax update
        const float mn = fmaxf(m, t);
        const float sc = __expf(m - mn);   // first iter: exp(-huge) -> 0
        const float w  = __expf(t - mn);
        l = l * sc + w;
        #pragma unroll
        for (int j = 0; j < 8; ++j) {
            acc[j].x = acc[j].x * sc + w * v[j].x;
            acc[j].y = acc[j].y * sc + w * v[j].y;
            acc[j].z = acc[j].z * sc + w * v[j].z;
            acc[j].w = acc[j].w * sc + w * v[j].w;
        }
        m = mn;
    }

    if (lane == 0) { wm[wave] = m; wl[wave] = l; }
    __syncthreads();

    // Block-level log-sum-exp merge of the 8 wave states
    float M = wm[0];
    #pragma unroll
    for (int c2 = 1; c2 < 8; ++c2) M = fmaxf(M, wm[c2]);
    float L = 0.0f;
    #pragma unroll
    for (int c2 = 0; c2 < 8; ++c2) L += wl[c2] * __expf(wm[c2] - M);

    const float sc2 = __expf(m - M);       // uniform within a wave
    #pragma unroll
    for (int j = 0; j < 8; ++j) {
        const int d = j * 128 + lane * 4;
        atomicAdd(&sacc[d + 0], acc[j].x * sc2);   // ds_add_f32
        atomicAdd(&sacc[d + 1], acc[j].y * sc2);
        atomicAdd(&sacc[d + 2], acc[j].z * sc2);
        atomicAdd(&sacc[d + 3], acc[j].w * sc2);
    }
    __syncthreads();

    const int pidx = b * 16 + chunk;
    if (threadIdx.x == 0) { partM[pidx] = M; partL[pidx] = L; }
    float* pc = partCtx + (size_t)pidx * DOUT_;
    for (int i = threadIdx.x; i < DOUT_; i += 256) pc[i] = sacc[i];
}

// ---------------------------------------------------------------------------
// Kernel 3: combine the 16 chunk partials per batch; write ctx + normalized
// attention probabilities.  grid = B, block = 256.
// ---------------------------------------------------------------------------
__global__ void __launch_bounds__(256)
attn_finalize_kernel(const float* __restrict__ raw_scores,
                     const float* __restrict__ partM,
                     const float* __restrict__ partL,
                     const float* __restrict__ partCtx,
                     float* __restrict__ out_ctx,    // [B,DOUT]
                     float* __restrict__ out_attn)   // [B,S]
{
    const int b = blockIdx.x;

    float Mc[16], wc[16];
    float M = -3.0e38f;
    #pragma unroll
    for (int c = 0; c < 16; ++c) { Mc[c] = partM[b * 16 + c]; M = fmaxf(M, Mc[c]); }
    float L = 0.0f;
    #pragma unroll
    for (int c = 0; c < 16; ++c) { wc[c] = __expf(Mc[c] - M); L += partL[b * 16 + c] * wc[c]; }
    const float invL = 1.0f / L;

    for (int d = threadIdx.x; d < DOUT_; d += 256) {
        float sum = 0.0f;
        #pragma unroll
        for (int c = 0; c < 16; ++c)
            sum += wc[c] * partCtx[(size_t)(b * 16 + c) * DOUT_ + d];
        out_ctx[(size_t)b * DOUT_ + d] = sum * invL;
    }
    for (int s = threadIdx.x; s < S_; s += 256)
        out_attn[(size_t)b * S_ + s] =
            __expf(raw_scores[(size_t)b * S_ + s] - M) * invL;
}

// ---------------------------------------------------------------------------
extern "C" void kernel_launch(void* const* d_in, const int* in_sizes, int n_in,
                              void* d_out, int out_size, void* d_ws, size_t ws_size,
                              hipStream_t stream) {
    const float* input       = (const float*)d_in[0];  // [B, DIN]
    const float* source_hids = (const float*)d_in[1];  // [B, S, DOUT]
    const float* seg_scores  = (const float*)d_in[2];  // [B, S]
    const float* W           = (const float*)d_in[3];  // [DOUT, DIN]

    float* out_ctx  = (float*)d_out;                       // [B, DOUT]
    float* out_attn = (float*)d_out + (size_t)B_ * DOUT_;  // [B, S]

    // Workspace layout (floats), total ~2.7 MB
    float* ws   = (float*)d_ws;
    float* xw   = ws;                                   // B*DOUT      = 32768
    float* raw  = xw  + (size_t)B_ * DOUT_;             // B*S         = 131072
    float* pM   = raw + (size_t)B_ * S_;                // B*16        = 512
    float* pL   = pM  + B_ * 16;                        // B*16        = 512
    float* pCtx = pL  + B_ * 16;                        // B*16*DOUT   = 524288

    // 1) x = input @ W^T via f32 WMMA
    xw_wmma_kernel<<<dim3(B_ / 16, DOUT_ / 16), 32, 0, stream>>>(input, W, xw);

    // 2) fused score + online softmax + partial context (single pass over HBM)
    attn_stream_kernel<<<dim3(B_, S_ / CHUNK), 256, 0, stream>>>(
        source_hids, seg_scores, xw, raw, pM, pL, pCtx);

    // 3) combine partials, emit ctx and normalized attention
    attn_finalize_kernel<<<B_, 256, 0, stream>>>(raw, pM, pL, pCtx,
                                                 out_ctx, out_attn);
}